// IOU3dLoss_56495999812217
// MI455X (gfx1250) — compile-verified
//
#include <hip/hip_runtime.h>
#include <math.h>

#define TPB   256
#define BOXF  7
#define PSTR  9   // per-thread polygon buffer stride (vertices); writes reach idx 8

#if defined(__has_builtin)
#  if __has_builtin(__builtin_amdgcn_global_load_async_to_lds_b32) && \
      __has_builtin(__builtin_amdgcn_s_wait_asynccnt)
#    define USE_ASYNC_LDS 1
#  endif
#endif

typedef __attribute__((address_space(1))) int GintAS1;
typedef __attribute__((address_space(3))) int LintAS3;

struct __align__(8) P2 { float x, y; };

// Clip convex polygon in[0..n) against half-plane  sgn*coord(axis) <= bound,
// inside iff f(p) = bound - sgn*coord >= 0.  Returns new vertex count.
// Branch-free body: both candidate vertices are stored unconditionally and the
// write cursor advances only for valid ones (stale slots get overwritten, and
// the padded stride-9 buffer absorbs the one possible uncounted write at idx 8).
__device__ __forceinline__ int clip_pass(const P2* __restrict__ in, int n,
                                         P2* __restrict__ out,
                                         int axis, float sgn, float bound)
{
    if (n == 0) return 0;
    int m = 0;
    P2 a = in[n - 1];
    float fa = bound - sgn * (axis ? a.y : a.x);
    for (int i = 0; i < n; ++i) {
        P2 b = in[i];
        float fb = bound - sgn * (axis ? b.y : b.x);
        bool ain = fa >= 0.0f;
        bool bin = fb >= 0.0f;
        // crossing point (garbage inf/nan when !crossing -> never counted/read)
        float t = __fdividef(fa, fa - fb);
        P2 p;
        p.x = fmaf(t, b.x - a.x, a.x);
        p.y = fmaf(t, b.y - a.y, a.y);
        out[m] = p;
        m += (ain != bin) ? 1 : 0;
        out[m] = b;
        m += bin ? 1 : 0;
        a = b; fa = fb;
    }
    return m;
}

__global__ __launch_bounds__(TPB)
void iou3d_loss_kernel(const float* __restrict__ inp,
                       const float* __restrict__ tgt,
                       const float* __restrict__ wgt,
                       float* __restrict__ out, int n)
{
    // Staging: coalesced async b32 streams, conflict-free reads (stride 7, gcd(7,64)=1)
    __shared__ float sIn[TPB * BOXF];
    __shared__ float sTg[TPB * BOXF];
    // Ping-pong polygon buffers: stride-9 float2 => bank (18t+2i)%64 injective over a wave
    __shared__ P2 sPoly[2][TPB * PSTR];

    const int tid   = threadIdx.x;
    const int i     = blockIdx.x * TPB + tid;
    const int gbase = blockIdx.x * TPB * BOXF;
    const int gtot  = n * BOXF;

#pragma unroll
    for (int k = tid; k < TPB * BOXF; k += TPB) {
        int g = gbase + k;
        if (g < gtot) {
#ifdef USE_ASYNC_LDS
            __builtin_amdgcn_global_load_async_to_lds_b32(
                (GintAS1*)(void*)(inp + g), (LintAS3*)(&sIn[k]), 0, 0);
            __builtin_amdgcn_global_load_async_to_lds_b32(
                (GintAS1*)(void*)(tgt + g), (LintAS3*)(&sTg[k]), 0, 0);
#else
            sIn[k] = inp[g];
            sTg[k] = tgt[g];
#endif
        }
    }
#ifdef USE_ASYNC_LDS
    __builtin_amdgcn_s_wait_asynccnt(0);
#endif
    __syncthreads();

    if (i >= n) return;

    // box layout: [cx, cy, cz, dx, dy, dz, heading]
    float b1[BOXF], b2[BOXF];
#pragma unroll
    for (int j = 0; j < BOXF; ++j) {
        float a = sIn[tid * BOXF + j];
        float t = sTg[tid * BOXF + j];
        b1[j] = a;
        b2[j] = (t != t) ? a : t;          // NaN target -> input
    }

    const float hx1 = b1[3] * 0.5f, hy1 = b1[4] * 0.5f;
    const float hx2 = b2[3] * 0.5f, hy2 = b2[4] * 0.5f;

    // Work in box1's local frame: box1 becomes [-hx1,hx1]x[-hy1,hy1].
    float s1, c1, sd, cd;
    __sincosf(b1[6], &s1, &c1);
    __sincosf(b2[6] - b1[6], &sd, &cd);
    const float dx = b2[0] - b1[0];
    const float dy = b2[1] - b1[1];
    const float tx = fmaf(dx, c1,  dy * s1);
    const float ty = fmaf(dy, c1, -dx * s1);

    P2* buf0 = &sPoly[0][tid * PSTR];
    P2* buf1 = &sPoly[1][tid * PSTR];

    // CCW corners of box2 in box1's frame
    const float sxs[4] = { 1.f, -1.f, -1.f,  1.f };
    const float sys[4] = { 1.f,  1.f, -1.f, -1.f };
#pragma unroll
    for (int k = 0; k < 4; ++k) {
        float lx = sxs[k] * hx2;
        float ly = sys[k] * hy2;
        P2 q;
        q.x = fmaf(lx, cd, fmaf(-ly, sd, tx));
        q.y = fmaf(lx, sd, fmaf( ly, cd, ty));
        buf0[k] = q;
    }

    // Sutherland–Hodgman against 4 axis-aligned half-planes of box1
    int m = 4;                                     // pass 1 trip count is static
    m = clip_pass(buf0, m, buf1, 0, -1.0f, hx1);   // x >= -hx1
    m = clip_pass(buf1, m, buf0, 0,  1.0f, hx1);   // x <=  hx1
    m = clip_pass(buf0, m, buf1, 1, -1.0f, hy1);   // y >= -hy1
    m = clip_pass(buf1, m, buf0, 1,  1.0f, hy1);   // y <=  hy1

    // Shoelace
    float area2 = 0.0f;
    if (m >= 3) {
        P2 a = buf0[m - 1];
        for (int k = 0; k < m; ++k) {
            P2 b = buf0[k];
            area2 = fmaf(a.x, b.y, area2);
            area2 = fmaf(-a.y, b.x, area2);
            a = b;
        }
    }
    const float bev = 0.5f * fabsf(area2);

    // z overlap + IoU3D
    const float zmax = fminf(fmaf(b1[5], 0.5f, b1[2]), fmaf(b2[5], 0.5f, b2[2]));
    const float zmin = fmaxf(fmaf(b1[5], -0.5f, b1[2]), fmaf(b2[5], -0.5f, b2[2]));
    const float inter = bev * fmaxf(zmax - zmin, 0.0f);
    const float vol1 = b1[3] * b1[4] * b1[5];
    const float vol2 = b2[3] * b2[4] * b2[5];
    const float iou  = __fdividef(inter, fmaxf(vol1 + vol2 - inter, 1e-7f));

    out[i] = (1.0f - iou) * wgt[i];
}

extern "C" void kernel_launch(void* const* d_in, const int* in_sizes, int n_in,
                              void* d_out, int out_size, void* d_ws, size_t ws_size,
                              hipStream_t stream)
{
    (void)n_in; (void)d_ws; (void)ws_size; (void)out_size;
    const float* inp = (const float*)d_in[0];
    const float* tgt = (const float*)d_in[1];
    const float* wgt = (const float*)d_in[2];
    float* out = (float*)d_out;
    const int n = in_sizes[0] / BOXF;
    const int blocks = (n + TPB - 1) / TPB;
    hipLaunchKernelGGL(iou3d_loss_kernel, dim3(blocks), dim3(TPB), 0, stream,
                       inp, tgt, wgt, out, n);
}